// DSQGAttentionV4_44023414784194
// MI455X (gfx1250) — compile-verified
//
#include <hip/hip_runtime.h>

typedef __attribute__((ext_vector_type(2))) float v2f;
typedef __attribute__((ext_vector_type(8))) float v8f;

#define NB 2
#define NH 16
#define NN 8192
#define HDIM 64
#define NOFF 44
#define NTILE 14
#define SSTR 52   // LDS score-row stride: 16 rows * 52 -> 16 distinct banks

__constant__ int c_off[NOFF] = {
  0,1,2,3,4,5,6,7,8,9,10,11,12,13,14,15,16,17,18,19,20,21,22,
  23,24,25,26,27,28,29,30,31,32,
  48,64,96,128,192,256,384,512,768,1024,1536 };

// column-tile base offsets (n0 - m0); tiles 0..2 dense band, 3..13 sparse diagonals
__constant__ int c_toff[NTILE] = {0,16,32,48,64,96,128,192,256,384,512,768,1024,1536};

__device__ __forceinline__ v8f wmma4(v2f a, v2f b, v8f c) {
  // D = A(16x4 f32) * B(4x16 f32) + C(16x16 f32)
  return __builtin_amdgcn_wmma_f32_16x16x4_f32(
      false, a, false, b, (short)0, c, false, false);
}

__global__ __launch_bounds__(128)
void dsqg_attn_v4_kernel(const float* __restrict__ q,
                         const float* __restrict__ k,
                         const float* __restrict__ v,
                         const float* __restrict__ pbias,
                         const float* __restrict__ se,
                         const float* __restrict__ ph,
                         float* __restrict__ out)
{
  __shared__ float s_cp[48], s_sp[48], s_pb[48];
  __shared__ float s_scores[4 * 16 * SSTR];
  __shared__ float s_rinv[4 * 16];

  const int bid  = blockIdx.x;
  const int bh   = bid >> 7;            // 128 row-blocks per (b,h)
  const int h    = bh & (NH - 1);
  const int n0b  = (bid & 127) << 6;    // 64 rows per block (4 waves x 16)
  const int tid  = threadIdx.x;
  const int wv   = tid >> 5;
  const int lane = tid & 31;
  const int n0   = n0b + (wv << 4);

  // per-head tables (cos/sin of phase, pos_bias)
  if (tid < NOFF) {
    float p = ph[tid * NH + h];
    s_cp[tid] = cosf(p);
    s_sp[tid] = sinf(p);
    s_pb[tid] = pbias[tid * NH + h];
  }
  __syncthreads();

  const float sc   = 0.125f;            // 1/sqrt(64)
  const int   nl   = lane & 15;
  const int   hi   = lane >> 4;
  const int   koff = hi << 1;           // K sub-offset per half-wave (A/B layout)

  const size_t bhoff = (size_t)bh * NN * HDIM;
  const float* qbh = q + bhoff;
  const float* kbh = k + bhoff;
  const float* vbh = v + bhoff;
  float*       obh = out + bhoff;
  float*       srow = s_scores + wv * (16 * SSTR);

  // ---------------- Q tile in WMMA A-layout (16x64), pre-scaled by sc --------
  // A 16x4 f32: lanes 0-15 -> M=lane (K=4c+0/1), lanes 16-31 -> M=lane-16 (K=4c+2/3)
  v2f qa[16];
  {
    const float* qr = qbh + (size_t)(n0 + nl) * HDIM + koff;
#pragma unroll
    for (int c = 0; c < 16; ++c) {
      v2f t = *(const v2f*)(qr + 4 * c);
      qa[c].x = t.x * sc;
      qa[c].y = t.y * sc;
    }
  }

  // ---------------- qdyn = (sc*Q) @ SE^T  -> initializes score LDS ----------
#pragma unroll
  for (int it = 0; it < 3; ++it) {
    int irow = it * 16 + nl;
    int ir   = (irow < NOFF) ? irow : (NOFF - 1);   // clamp; garbage cols unused
    const float* ser = se + (size_t)ir * HDIM + koff;
    v8f acc = {0.f,0.f,0.f,0.f,0.f,0.f,0.f,0.f};
#pragma unroll
    for (int c = 0; c < 16; ++c) {
      v2f b = *(const v2f*)(ser + 4 * c);
      acc = wmma4(qa[c], b, acc);
    }
#pragma unroll
    for (int r = 0; r < 8; ++r) {
      int M = r + (hi << 3);
      int i = it * 16 + nl;
      if (i < NOFF) srow[M * SSTR + i] = acc[r];
    }
  }

  // ---------------- banded / sparse-diagonal Q.K^T tiles --------------------
  for (int t = 0; t < NTILE; ++t) {
    const int off = c_toff[t];
    if (n0 + 15 < off) continue;        // wave-uniform: whole tile causally masked
    const int m0 = n0 - off;
    int krow = m0 + nl; if (krow < 0) krow = 0;   // clamped rows masked later
    const float* kr = kbh + (size_t)krow * HDIM + koff;
    __builtin_prefetch(vbh + (size_t)krow * HDIM, 0, 1);  // warm V for output pass
    v8f acc = {0.f,0.f,0.f,0.f,0.f,0.f,0.f,0.f};
#pragma unroll
    for (int c = 0; c < 16; ++c) {
      v2f b = *(const v2f*)(kr + 4 * c);
      acc = wmma4(qa[c], b, acc);
    }
    // C layout: VGPR r, lanes0-15 -> (M=r, N=lane); lanes16-31 -> (M=8+r, N=lane-16)
    if (t < 3) {                        // dense band: d = M - N + 16t, keep 0..32
#pragma unroll
      for (int r = 0; r < 8; ++r) {
        int M = r + (hi << 3);
        int d = M - nl + (t << 4);
        if (d >= 0 && d <= 32) atomicAdd(&srow[M * SSTR + d], acc[r]); // ds_add_f32
      }
    } else {                            // sparse: only main diagonal M==N
      const int i = 33 + (t - 3);
#pragma unroll
      for (int r = 0; r < 8; ++r) {
        int M = r + (hi << 3);
        if (M == nl) atomicAdd(&srow[M * SSTR + i], acc[r]);
      }
    }
  }

  // ---------------- softmax per row (lanes 0..15 own one row each) ----------
  if (lane < 16) {
    float* sr = srow + lane * SSTR;
    const int ng = n0 + lane;
    float mx = -3.0e38f;
    for (int i = 0; i < NOFF; ++i) {
      float sv = sr[i] + s_pb[i];
      if (ng < c_off[i]) sv = -3.0e38f;   // causal mask -> exp == 0
      sr[i] = sv;
      mx = fmaxf(mx, sv);
    }
    float sum = 0.f;
    for (int i = 0; i < NOFF; ++i) {
      float e = __expf(sr[i] - mx);
      sr[i] = e;                          // unnormalized p
      sum += e;
    }
    s_rinv[wv * 16 + lane] = 1.0f / sum;
  }

  const float rinv = s_rinv[wv * 16 + nl];

  // ---------------- output: out = Pc.V + J(Ps.V) ----------------------------
  v8f oc[4], osn[4];
#pragma unroll
  for (int ht = 0; ht < 4; ++ht) {
    oc[ht]  = (v8f){0.f,0.f,0.f,0.f,0.f,0.f,0.f,0.f};
    osn[ht] = (v8f){0.f,0.f,0.f,0.f,0.f,0.f,0.f,0.f};
  }

  for (int t = 0; t < NTILE; ++t) {
    const int off = c_toff[t];
    if (n0 + 15 < off) continue;
    const int m0 = n0 - off;

    // build Pc/Ps A-operands (16x16, K = key position within tile)
    v2f pac[4], pas[4];
#pragma unroll
    for (int c = 0; c < 4; ++c) {
#pragma unroll
      for (int u = 0; u < 2; ++u) {
        int j = 4 * c + koff + u;         // K index (M = nl for both half-waves)
        int iidx = 0; bool valid;
        if (t < 3) { int d = nl - j + (t << 4); valid = (d >= 0) && (d <= 32); iidx = d; }
        else       { valid = (nl == j); iidx = 33 + (t - 3); }
        float pc = 0.f, ps = 0.f;
        if (valid) {
          float pv = srow[nl * SSTR + iidx] * rinv;
          pc = pv * s_cp[iidx];
          ps = pv * s_sp[iidx];
        }
        if (u == 0) { pac[c].x = pc; pas[c].x = ps; }
        else        { pac[c].y = pc; pas[c].y = ps; }
      }
    }

    // V in B-layout; Pc and Ps accumulators share the same B operands
#pragma unroll
    for (int ht = 0; ht < 4; ++ht) {
      const int hd0 = ht << 4;
#pragma unroll
      for (int c = 0; c < 4; ++c) {
        int rr = m0 + 4 * c + koff;
        int r0 = rr < 0 ? 0 : rr;
        int r1 = (rr + 1) < 0 ? 0 : (rr + 1);
        v2f vb;
        vb.x = vbh[(size_t)r0 * HDIM + hd0 + nl];
        vb.y = vbh[(size_t)r1 * HDIM + hd0 + nl];
        oc[ht]  = wmma4(pac[c], vb, oc[ht]);
        osn[ht] = wmma4(pas[c], vb, osn[ht]);
      }
    }
  }

  // ---------------- apply J to sin-part columns and store -------------------
  // out[.,e]   = PcV[.,e]   - PsV[.,e+1]   (e even)
  // out[.,e+1] = PcV[.,e+1] + PsV[.,e]
  const float sgn = (lane & 1) ? 1.0f : -1.0f;
#pragma unroll
  for (int ht = 0; ht < 4; ++ht) {
#pragma unroll
    for (int r = 0; r < 8; ++r) {
      float sv = osn[ht][r];
      float sw = __shfl_xor(sv, 1, 32);   // neighbor column (pair partner)
      float res = oc[ht][r] + sgn * sw;
      int M = r + (hi << 3);
      obh[(size_t)(n0 + M) * HDIM + (ht << 4) + nl] = res;
    }
  }
}

extern "C" void kernel_launch(void* const* d_in, const int* in_sizes, int n_in,
                              void* d_out, int out_size, void* d_ws, size_t ws_size,
                              hipStream_t stream) {
  const float* q  = (const float*)d_in[0];
  const float* k  = (const float*)d_in[1];
  const float* v  = (const float*)d_in[2];
  const float* pb = (const float*)d_in[3];
  const float* se = (const float*)d_in[4];
  const float* ph = (const float*)d_in[5];
  float* out = (float*)d_out;

  dim3 grid(NB * NH * (NN / 64));   // 4096 blocks; block = 4 waves x 16 rows
  dim3 block(128);
  hipLaunchKernelGGL(dsqg_attn_v4_kernel, grid, block, 0, stream,
                     q, k, v, pb, se, ph, out);
}